// QIDDM_L_B_21285857919137
// MI455X (gfx1250) — compile-verified
//
#include <hip/hip_runtime.h>
#include <hip/hip_bf16.h>

typedef float v2f __attribute__((ext_vector_type(2)));
typedef float v8f __attribute__((ext_vector_type(8)));

#define HF 14
#define DIM 16384          // 2^14 amplitudes
#define SPECTRUM 3
#define SEL_L 3
#define NSTEPS 2
#define BATCH 64
#define IN_DIM 784
#define XR_LD 16           // padded row stride for the (64 x 14) activations
#define CTH 1024           // circuit threads per workgroup (32 waves = 1 WGP)
#define PI_F 3.14159265358979323846f

__device__ inline v8f wmma4(v2f a, v2f b, v8f c) {
    return __builtin_amdgcn_wmma_f32_16x16x4_f32(false, a, false, b,
                                                 (short)0, c, false, false);
}

// ---------------------------------------------------------------------------
// Rot(phi,theta,omega) = RZ(omega) RY(theta) RZ(phi)  -- one complex entry
// ---------------------------------------------------------------------------
__device__ inline void rot_entry(float phi, float theta, float omega,
                                 int r, int c, float& re, float& im) {
    float ch = __cosf(0.5f * theta);
    float sh = __sinf(0.5f * theta);
    float ang, mag;
    if (r == 0 && c == 0)      { ang = -0.5f * (phi + omega); mag =  ch; }
    else if (r == 0 && c == 1) { ang =  0.5f * (phi - omega); mag = -sh; }
    else if (r == 1 && c == 0) { ang = -0.5f * (phi - omega); mag =  sh; }
    else                       { ang =  0.5f * (phi + omega); mag =  ch; }
    float sa, ca;
    __sincosf(ang, &sa, &ca);
    re = mag * ca;
    im = mag * sa;
}

// ---------------------------------------------------------------------------
// linear_down: xr(64x16,padded) = x(64x784) @ w_down^T(784x14) + b_down
// ---------------------------------------------------------------------------
__global__ __launch_bounds__(32)
void lin_down_kernel(const float* __restrict__ x, const float* __restrict__ wd,
                     const float* __restrict__ bd, float* __restrict__ xr) {
    const int lane  = threadIdx.x & 31;
    const int mt    = blockIdx.x;            // 0..3
    const int n     = lane & 15;
    const int m     = mt * 16 + n;
    const int khalf = (lane >> 4) << 1;      // K sub-offset 0 or 2
    // clamped row: garbage columns n>=14 are zeroed at store time
    const float* __restrict__ arow = x  + m * IN_DIM;
    const float* __restrict__ brow = wd + (n < HF ? n : 0) * IN_DIM;
    v8f acc = {};
    for (int k0 = 0; k0 < IN_DIM; k0 += 4) {
        const int ka = k0 + khalf;
        v2f a, b;
        a.x = arow[ka];     a.y = arow[ka + 1];
        b.x = brow[ka];     b.y = brow[ka + 1];
        acc = wmma4(a, b, acc);
    }
    const int mrow = (lane >> 4) * 8;
    #pragma unroll
    for (int r = 0; r < 8; ++r) {
        const int mm = mt * 16 + mrow + r;
        xr[mm * XR_LD + n] = (n < HF) ? (acc[r] + bd[n]) : 0.f;
    }
}

// ---------------------------------------------------------------------------
// BatchNorm1d (training-mode batch stats) over (64 x 14), in place.
// ---------------------------------------------------------------------------
__global__ __launch_bounds__(32)
void bn_kernel(float* __restrict__ v, const float* __restrict__ gw,
               const float* __restrict__ gb) {
    const int j = threadIdx.x;
    if (j >= HF) return;
    float mu = 0.f;
    for (int b = 0; b < BATCH; ++b) mu += v[b * XR_LD + j];
    mu *= (1.f / BATCH);
    float var = 0.f;
    for (int b = 0; b < BATCH; ++b) {
        const float d = v[b * XR_LD + j] - mu;
        var += d * d;
    }
    var *= (1.f / BATCH);
    const float sc = rsqrtf(var + 1e-5f) * gw[j];
    const float of = gb[j];
    for (int b = 0; b < BATCH; ++b)
        v[b * XR_LD + j] = (v[b * XR_LD + j] - mu) * sc + of;
}

// ---------------------------------------------------------------------------
// Build 16x16 complex kron unitary (one entry per thread, threads 0..255).
// ---------------------------------------------------------------------------
__device__ inline void build_U(const float* __restrict__ wl, int w0, bool idhalf,
                               int tid, float (&Ur)[16][16], float (&Ui)[16][16]) {
    if (tid >= 256) return;
    const int a = tid >> 4, c = tid & 15;
    float pr = 1.f, pi = 0.f;
    #pragma unroll
    for (int t = 0; t < 4; ++t) {
        const int at = (a >> (3 - t)) & 1;
        const int ct = (c >> (3 - t)) & 1;
        float er, ei;
        if (idhalf && t < 2) {                 // identity padding (wires 10,11)
            er = (at == ct) ? 1.f : 0.f; ei = 0.f;
        } else {
            const float* wq = wl + (w0 + t) * 3;
            rot_entry(wq[0], wq[1], wq[2], at, ct, er, ei);
        }
        const float nr = pr * er - pi * ei;
        const float ni = pr * ei + pi * er;
        pr = nr; pi = ni;
    }
    Ur[a][c] = pr;
    Ui[a][c] = pi;
}

// ---------------------------------------------------------------------------
// Apply 16x16 complex unitary on the 4-wire group starting at wire W0.
// State viewed as (pre, 16, POST); 1024 columns, 16 per wave, 2 blocks/wave.
// ---------------------------------------------------------------------------
template <int W0>
__device__ inline void apply_group(float2* __restrict__ st,
                                   const float (&Ur)[16][16],
                                   const float (&Ui)[16][16],
                                   int lane, int wave) {
    constexpr int POST = 1 << (10 - W0);
    const int n     = lane & 15;
    const int khalf = (lane >> 4) << 1;
    const int mbase = (lane >> 4) * 8;

    // hoist loop-invariant A-side (U) fragments
    v2f aur[4], aui[4], anui[4];
    #pragma unroll
    for (int kc = 0; kc < 4; ++kc) {
        const int k0 = kc * 4 + khalf;
        aur[kc].x  =  Ur[n][k0];     aur[kc].y  =  Ur[n][k0 + 1];
        aui[kc].x  =  Ui[n][k0];     aui[kc].y  =  Ui[n][k0 + 1];
        anui[kc].x = -aui[kc].x;     anui[kc].y = -aui[kc].y;
    }

    #pragma unroll
    for (int blk = wave; blk < 64; blk += 32) {
        const int col  = blk * 16 + n;
        const int p    = col / POST;            // compile-time shift
        const int q    = col % POST;
        const int base = p * (16 * POST) + q;
        v8f dr = {}, di = {};
        #pragma unroll
        for (int kc = 0; kc < 4; ++kc) {
            const int k0 = kc * 4 + khalf;
            const float2 s0 = st[base + k0 * POST];
            const float2 s1 = st[base + (k0 + 1) * POST];
            v2f br, bi;
            br.x = s0.x; bi.x = s0.y;
            br.y = s1.x; bi.y = s1.y;
            // Dr += Ur*Sr - Ui*Si ; Di += Ur*Si + Ui*Sr
            dr = wmma4(aur[kc],  br, dr);
            dr = wmma4(anui[kc], bi, dr);
            di = wmma4(aur[kc],  bi, di);
            di = wmma4(aui[kc],  br, di);
        }
        #pragma unroll
        for (int r = 0; r < 8; ++r) {
            float2 o; o.x = dr[r]; o.y = di[r];
            st[base + (mbase + r) * POST] = o;
        }
    }
}

// ---------------------------------------------------------------------------
// Quantum circuit: one full-WGP workgroup (32 waves) per batch element;
// 16384-amp state lives in LDS for the whole circuit.
// ---------------------------------------------------------------------------
__global__ __launch_bounds__(CTH)
void circuit_kernel(const float* __restrict__ xin,   // (64 x 16) angles
                    const float* __restrict__ wts,   // (SPECTRUM,SEL_L,HF,3)
                    float* __restrict__ xout) {      // (64 x 16) expvals
    extern __shared__ float state[];                 // 2*DIM floats
    __shared__ float Ur[16][16];
    __shared__ float Ui[16][16];
    __shared__ float ang[16];
    __shared__ float phhi[128];                      // wires 0..6  (s >> 7)
    __shared__ float phlo[128];                      // wires 7..13 (s & 127)
    __shared__ unsigned czb[SEL_L][DIM / 32];        // packed CZ sign bits
    __shared__ float red[32];

    const int tid  = threadIdx.x;
    const int b    = blockIdx.x;
    const int lane = tid & 31;
    const int wave = tid >> 5;                       // 0..31
    float2* st = (float2*)state;

    if (tid == 0) __builtin_prefetch(wts, 0, 0);     // global_prefetch_b8
    if (tid < 16) ang[tid] = xin[b * XR_LD + tid];
    for (int s = tid; s < DIM; s += CTH) {
        float2 z; z.x = (s == 0) ? 1.f : 0.f; z.y = 0.f;
        st[s] = z;
    }
    __syncthreads();

    // --- one-time tables: split RZ phase, packed CZ ring signs --------------
    if (tid < 128) {
        float p = 0.f;
        #pragma unroll
        for (int j = 0; j < 7; ++j)
            p += ang[j] * (((tid >> (6 - j)) & 1) - 0.5f);
        phhi[tid] = p;
    } else if (tid < 256) {
        const int h = tid - 128;
        float p = 0.f;
        #pragma unroll
        for (int j = 7; j < HF; ++j)
            p += ang[j] * (((h >> (HF - 1 - j)) & 1) - 0.5f);
        phlo[h] = p;
    }
    for (int wix = tid; wix < SEL_L * (DIM / 32); wix += CTH) {
        const int l  = wix >> 9;                     // / 512
        const int w  = wix & 511;
        const int rr = l + 1;
        unsigned bits = 0u;
        for (int t = 0; t < 32; ++t) {
            const int s = (w << 5) | t;
            int par = 0;
            #pragma unroll
            for (int j = 0; j < HF; ++j)
                par ^= ((s >> (HF - 1 - j)) & (s >> (HF - 1 - ((j + rr) % HF)))) & 1;
            bits |= (unsigned)par << t;
        }
        czb[l][w] = bits;
    }
    __syncthreads();

    for (int spec = 0; spec < SPECTRUM; ++spec) {
        // --- RZ diagonal; for spec>0 fuse the previous layer's l=2 CZ sign
        //     as a +pi phase offset (diagonals compose) ----------------------
        const bool fuse_cz = (spec > 0);
        for (int s = tid; s < DIM; s += CTH) {
            float ph = phhi[s >> 7] + phlo[s & 127];
            if (fuse_cz && ((czb[SEL_L - 1][s >> 5] >> (s & 31)) & 1u))
                ph += PI_F;                          // e^{i(ph+pi)} = -e^{i ph}
            float sp, cp;
            __sincosf(ph, &sp, &cp);
            const float2 a = st[s];
            float2 o;
            o.x = a.x * cp - a.y * sp;
            o.y = a.x * sp + a.y * cp;
            st[s] = o;
        }
        __syncthreads();

        for (int l = 0; l < SEL_L; ++l) {
            const float* wl = wts + (spec * SEL_L + l) * HF * 3;

            build_U(wl, 0, false, tid, Ur, Ui);
            __syncthreads();
            apply_group<0>(st, Ur, Ui, lane, wave);
            __syncthreads();

            build_U(wl, 4, false, tid, Ur, Ui);
            __syncthreads();
            apply_group<4>(st, Ur, Ui, lane, wave);
            __syncthreads();

            build_U(wl, 8, false, tid, Ur, Ui);
            __syncthreads();
            apply_group<8>(st, Ur, Ui, lane, wave);
            __syncthreads();

            build_U(wl, 10, true, tid, Ur, Ui);      // I x I x R12 x R13
            __syncthreads();
            apply_group<10>(st, Ur, Ui, lane, wave);
            __syncthreads();

            // --- CZ ring diagonal: only needed standalone for l<2.
            //     l=2 sign is fused into next RZ (spec<2) or dropped (spec=2,
            //     a global +-1 diagonal right before |amp|^2 is unobservable).
            if (l < SEL_L - 1) {
                for (int s = tid; s < DIM; s += CTH) {
                    if ((czb[l][s >> 5] >> (s & 31)) & 1u) {
                        float2 a = st[s];
                        a.x = -a.x; a.y = -a.y;
                        st[s] = a;
                    }
                }
                __syncthreads();
            }
        }
    }

    // --- <Z_j> = sum_s |amp_s|^2 * (1 - 2 b_j(s)) ---------------------------
    float loc[HF];
    #pragma unroll
    for (int j = 0; j < HF; ++j) loc[j] = 0.f;
    for (int s = tid; s < DIM; s += CTH) {
        const float2 a = st[s];
        const float p = a.x * a.x + a.y * a.y;
        #pragma unroll
        for (int j = 0; j < HF; ++j)
            loc[j] += ((s >> (HF - 1 - j)) & 1) ? -p : p;
    }
    for (int j = 0; j < HF; ++j) {
        float v = loc[j];
        #pragma unroll
        for (int off = 16; off > 0; off >>= 1)
            v += __shfl_down(v, off, 32);
        if (lane == 0) red[wave] = v;
        __syncthreads();
        if (tid < 32) {
            float t = red[tid];
            #pragma unroll
            for (int off = 16; off > 0; off >>= 1)
                t += __shfl_down(t, off, 32);
            if (tid == 0) xout[b * XR_LD + j] = t;
        }
        __syncthreads();
    }
    if (tid == 0) { xout[b * XR_LD + 14] = 0.f; xout[b * XR_LD + 15] = 0.f; }
}

// ---------------------------------------------------------------------------
// linear_up: out(64x784) = xr(64x16,padded) @ w_up^T(14x784) + b_up
// ---------------------------------------------------------------------------
__global__ __launch_bounds__(32)
void lin_up_kernel(const float* __restrict__ xr, const float* __restrict__ wu,
                   const float* __restrict__ bu, float* __restrict__ out) {
    const int lane  = threadIdx.x & 31;
    const int mt    = blockIdx.x & 3;        // 4 M-tiles
    const int nt    = blockIdx.x >> 2;       // 49 N-tiles
    const int idx   = lane & 15;
    const int m     = mt * 16 + idx;
    const int n     = nt * 16 + idx;
    const int khalf = (lane >> 4) << 1;
    v8f acc = {};
    #pragma unroll
    for (int kc = 0; kc < 4; ++kc) {
        const int k  = kc * 4 + khalf;
        // A-side padding (xr cols 14,15) is exact zero, so clamped B garbage
        // contributes 0 * finite = 0; clamp keeps loads in-bounds.
        const int k0 = (k     < HF) ? k     : (HF - 1);
        const int k1 = (k + 1 < HF) ? k + 1 : (HF - 1);
        v2f a, b;
        a.x = xr[m * XR_LD + k];
        a.y = xr[m * XR_LD + k + 1];
        b.x = wu[n * HF + k0];
        b.y = wu[n * HF + k1];
        acc = wmma4(a, b, acc);
    }
    const int mrow = (lane >> 4) * 8;
    #pragma unroll
    for (int r = 0; r < 8; ++r) {
        const int mm = mt * 16 + mrow + r;
        out[mm * IN_DIM + n] = acc[r] + bu[n];
    }
}

// ---------------------------------------------------------------------------
extern "C" void kernel_launch(void* const* d_in, const int* in_sizes, int n_in,
                              void* d_out, int out_size, void* d_ws, size_t ws_size,
                              hipStream_t stream) {
    const float* x   = (const float*)d_in[0];   // (64,1,28,28)
    const float* wd  = (const float*)d_in[1];   // (14,784)
    const float* bd  = (const float*)d_in[2];   // (14)
    const float* bnw = (const float*)d_in[3];   // (14)
    const float* bnb = (const float*)d_in[4];   // (14)
    const float* wq  = (const float*)d_in[5];   // (2,3,3,14,3)
    const float* wu  = (const float*)d_in[6];   // (784,14)
    const float* bu  = (const float*)d_in[7];   // (784)
    float* out = (float*)d_out;

    float* xrA = (float*)d_ws;                  // (64 x 16) padded
    float* xrB = xrA + BATCH * XR_LD;           // (64 x 16) padded

    lin_down_kernel<<<4, 32, 0, stream>>>(x, wd, bd, xrA);

    const size_t lds_bytes = (size_t)2 * DIM * sizeof(float);   // 128 KB state
    for (int n = 0; n < NSTEPS; ++n) {
        float* cur = (n == 0) ? xrA : xrB;
        float* nxt = (n == 0) ? xrB : xrA;
        bn_kernel<<<1, 32, 0, stream>>>(cur, bnw, bnb);
        circuit_kernel<<<BATCH, CTH, lds_bytes, stream>>>(
            cur, wq + (size_t)n * SPECTRUM * SEL_L * HF * 3, nxt);
    }

    lin_up_kernel<<<196, 32, 0, stream>>>(xrA, wu, bu, out);
}